// GNNTransformer_73263552135827
// MI455X (gfx1250) — compile-verified
//
#include <hip/hip_runtime.h>
#include <hip/hip_bf16.h>

typedef __attribute__((ext_vector_type(2))) float v2f;
typedef __attribute__((ext_vector_type(8))) float v8f;

#define IN_C 128
#define HID 64
#define NG 256
#define OUT_C 8

// ---------------- degree / normalization ----------------
__global__ __launch_bounds__(256) void deg_init(float* deg, int n) {
    int i = blockIdx.x * 256 + threadIdx.x;
    if (i < n) deg[i] = 1.0f;  // self-loop contributes 1
}

__global__ __launch_bounds__(256) void deg_count(const int* __restrict__ col,
                                                 float* deg, int e) {
    int i = blockIdx.x * 256 + threadIdx.x;
    if (i < e) atomicAdd(&deg[col[i]], 1.0f);
}

__global__ __launch_bounds__(256) void deg_rsqrt(float* deg, int n) {
    int i = blockIdx.x * 256 + threadIdx.x;
    if (i < n) {
        float d = deg[i];
        deg[i] = (d > 0.0f) ? rsqrtf(d) : 0.0f;   // deg >= 1 always, keep ref semantics
    }
}

// ---------------- x @ W via V_WMMA_F32_16X16X4_F32 ----------------
// One wave per 16-row M-tile; 4 N-tiles of 16 (HID=64); K=128 in steps of 4.
__global__ __launch_bounds__(256) void gemm_xw_wmma(const float* __restrict__ x,
                                                    const float* __restrict__ W,
                                                    float* __restrict__ xlin,
                                                    int M, int ntiles) {
    int wave = threadIdx.x >> 5;
    int lane = threadIdx.x & 31;
    int tile = blockIdx.x * 8 + wave;          // M-tile id (wave-uniform)
    if (tile >= ntiles) return;                // uniform exit: EXEC stays all-ones

    int mA   = tile * 16 + (lane & 15);        // A row for this lane
    if (mA >= M) mA = M - 1;                   // safety clamp (M % 16 == 0 in practice)
    int koff = (lane >> 4) << 1;               // 0 (lanes 0-15) or 2 (lanes 16-31)
    int nl   = lane & 15;                      // N within tile

    v8f acc0 = {}, acc1 = {}, acc2 = {}, acc3 = {};

    for (int k = 0; k < IN_C; k += 4) {
        const float* ap = x + (size_t)mA * IN_C + k + koff;
        v2f a; a.x = ap[0]; a.y = ap[1];

        const float* bp = W + (size_t)(k + koff) * HID + nl;
        v2f b0; b0.x = bp[0];      b0.y = bp[HID];
        v2f b1; b1.x = bp[16];     b1.y = bp[HID + 16];
        v2f b2; b2.x = bp[32];     b2.y = bp[HID + 32];
        v2f b3; b3.x = bp[48];     b3.y = bp[HID + 48];

        acc0 = __builtin_amdgcn_wmma_f32_16x16x4_f32(false, a, false, b0, (short)0, acc0, false, false);
        acc1 = __builtin_amdgcn_wmma_f32_16x16x4_f32(false, a, false, b1, (short)0, acc1, false, false);
        acc2 = __builtin_amdgcn_wmma_f32_16x16x4_f32(false, a, false, b2, (short)0, acc2, false, false);
        acc3 = __builtin_amdgcn_wmma_f32_16x16x4_f32(false, a, false, b3, (short)0, acc3, false, false);
    }

    // D layout: VGPR r -> M = tile*16 + r + 8*(lane>=16), N = lane&15
    int rbase = tile * 16 + ((lane >> 4) << 3);
#pragma unroll
    for (int r = 0; r < 8; ++r) {
        int m = rbase + r;
        if (m < M) {
            float* op = xlin + (size_t)m * HID + nl;
            op[0]  = acc0[r];
            op[16] = acc1[r];
            op[32] = acc2[r];
            op[48] = acc3[r];
        }
    }
}

// ---------------- self loop: h = xlin * dinv^2 ----------------
__global__ __launch_bounds__(256) void self_loop_init(const float* __restrict__ xlin,
                                                      const float* __restrict__ dinv,
                                                      float* __restrict__ h, int n) {
    int idx = blockIdx.x * 256 + threadIdx.x;   // over n*64
    int i = idx >> 6;
    if (i >= n) return;
    float di = dinv[i];
    h[idx] = xlin[idx] * di * di;
}

// ---------------- edge scatter: h[col] += xlin[row] * dinv[row]*dinv[col] ----------------
__global__ __launch_bounds__(256) void edge_scatter(const int* __restrict__ row,
                                                    const int* __restrict__ col,
                                                    const float* __restrict__ dinv,
                                                    const float* __restrict__ xlin,
                                                    float* __restrict__ h, int e) {
    int idx = blockIdx.x * 256 + threadIdx.x;   // over e*64
    int ei = idx >> 6;
    if (ei >= e) return;
    int c = idx & 63;
    int r = row[ei];
    int d = col[ei];
    float nrm = dinv[r] * dinv[d];
    atomicAdd(&h[(size_t)d * HID + c], xlin[(size_t)r * HID + c] * nrm);
}

// ---------------- bias + relu + pooled sums ----------------
__global__ __launch_bounds__(256) void relu_pool(const float* __restrict__ h,
                                                 const float* __restrict__ b,
                                                 const int* __restrict__ batch,
                                                 float* __restrict__ pooled, int n) {
    int idx = blockIdx.x * 256 + threadIdx.x;   // over n*64
    int i = idx >> 6;
    if (i >= n) return;
    int c = idx & 63;
    float v = h[idx] + b[c];
    v = v > 0.0f ? v : 0.0f;
    atomicAdd(&pooled[(size_t)batch[i] * HID + c], v);
}

__global__ __launch_bounds__(256) void count_nodes(const int* __restrict__ batch,
                                                   float* cnt, int n) {
    int i = blockIdx.x * 256 + threadIdx.x;
    if (i < n) atomicAdd(&cnt[batch[i]], 1.0f);
}

__global__ __launch_bounds__(256) void zero_f32(float* p, int n) {
    int i = blockIdx.x * 256 + threadIdx.x;
    if (i < n) p[i] = 0.0f;
}

// ---------------- final FC: out = (pooled/cnt) @ Wfc + bfc ----------------
__global__ __launch_bounds__(256) void fc_out(const float* __restrict__ pooled,
                                              const float* __restrict__ cnt,
                                              const float* __restrict__ Wfc,
                                              const float* __restrict__ bfc,
                                              float* __restrict__ out) {
    int idx = blockIdx.x * 256 + threadIdx.x;   // over NG*OUT_C = 2048
    if (idx >= NG * OUT_C) return;
    int g = idx >> 3;
    int o = idx & 7;
    float inv = 1.0f / fmaxf(cnt[g], 1.0f);
    float s = bfc[o];
#pragma unroll
    for (int c = 0; c < HID; ++c)
        s += pooled[g * HID + c] * inv * Wfc[c * OUT_C + o];
    out[idx] = s;
}

extern "C" void kernel_launch(void* const* d_in, const int* in_sizes, int n_in,
                              void* d_out, int out_size, void* d_ws, size_t ws_size,
                              hipStream_t stream) {
    const float* x     = (const float*)d_in[0];
    const int*   ei    = (const int*)d_in[1];
    const int*   batch = (const int*)d_in[2];
    const float* W     = (const float*)d_in[3];
    const float* b     = (const float*)d_in[4];
    const float* Wfc   = (const float*)d_in[5];
    const float* bfc   = (const float*)d_in[6];
    float* out = (float*)d_out;

    const int Nn = in_sizes[0] / IN_C;   // 50000
    const int E  = in_sizes[1] / 2;      // 800000
    const int* row = ei;
    const int* col = ei + E;

    // workspace carve-up (256B aligned regions)
    char* ws = (char*)d_ws;
    size_t o = 0;
    auto carve = [&](size_t bytes) -> char* {
        char* p = ws + o;
        o = (o + bytes + 255) & ~(size_t)255;
        return p;
    };
    float* dinv   = (float*)carve((size_t)Nn * 4);                 // deg -> dinv in place
    float* xlin   = (float*)carve((size_t)Nn * HID * 4);
    float* h      = (float*)carve((size_t)Nn * HID * 4);
    float* pooled = (float*)carve((size_t)(NG * HID + NG) * 4);    // pooled | cnt contiguous
    float* cnt    = pooled + NG * HID;
    (void)ws_size; (void)n_in; (void)out_size;

    const int T = 256;
    int nNodes   = Nn;
    int nNodeCh  = Nn * HID;            // 3.2M
    int nEdgeCh  = E * HID;             // 51.2M (fits int)

    zero_f32<<<(NG * HID + NG + T - 1) / T, T, 0, stream>>>(pooled, NG * HID + NG);
    deg_init<<<(nNodes + T - 1) / T, T, 0, stream>>>(dinv, Nn);
    deg_count<<<(E + T - 1) / T, T, 0, stream>>>(col, dinv, E);
    deg_rsqrt<<<(nNodes + T - 1) / T, T, 0, stream>>>(dinv, Nn);

    int ntiles = (Nn + 15) / 16;                       // 3125
    gemm_xw_wmma<<<(ntiles + 7) / 8, T, 0, stream>>>(x, W, xlin, Nn, ntiles);

    self_loop_init<<<(nNodeCh + T - 1) / T, T, 0, stream>>>(xlin, dinv, h, Nn);
    edge_scatter<<<(nEdgeCh + T - 1) / T, T, 0, stream>>>(row, col, dinv, xlin, h, E);
    relu_pool<<<(nNodeCh + T - 1) / T, T, 0, stream>>>(h, b, batch, pooled, Nn);
    count_nodes<<<(nNodes + T - 1) / T, T, 0, stream>>>(batch, cnt, Nn);
    fc_out<<<(NG * OUT_C + T - 1) / T, T, 0, stream>>>(pooled, cnt, Wfc, bfc, out);
}